// MosquitoDynamics_34024730919264
// MI455X (gfx1250) — compile-verified
//
#include <hip/hip_runtime.h>
#include <hip/hip_bf16.h>
#include <math.h>

typedef __attribute__((ext_vector_type(2))) float v2f;
typedef __attribute__((ext_vector_type(8))) float v8f;

// Branch-free tanh on raw HW transcendentals:
//   tanh(x) = 1 - 2/(exp(2x)+1)
// exp overflow (x large)  -> e=inf -> 1 - 0      = +1   (correct saturation)
// exp underflow (x small) -> e=0   -> 1 - 2/1    = -1   (correct saturation)
__device__ __forceinline__ float fast_tanhf(float x) {
    const float e = __expf(2.0f * x);                       // v_exp_f32
    const float r = __builtin_amdgcn_rcpf(e + 1.0f);        // v_rcp_f32
    return fmaf(-2.0f, r, 1.0f);
}

// ---------------------------------------------------------------------------
// Kernel 1: MLP via f32 WMMA (16x16x4, K=3 padded to 4) + per-month params.
// One wave -> 16 rows.  Block = 256 threads = 8 waves = 128 rows.
// Writes: ovi -> d_out[2n..3n)  and ws4[row] = {ovi, dp, ma, mp}
// ---------------------------------------------------------------------------
__global__ void __launch_bounds__(256)
mosq_mlp_params_kernel(const float* __restrict__ Xn,   // weather_norm (n,3)
                       const float* __restrict__ Xr,   // weather_raw  (n,3)
                       const float* __restrict__ W1,   // (3,64) row-major
                       const float* __restrict__ b1,   // (64)
                       const float* __restrict__ W2,   // (64,1)
                       const float* __restrict__ b2,   // (1)
                       float* __restrict__ ovi_out,    // d_out + 2n
                       float* __restrict__ ws4,        // (n,4) scratch
                       int n)
{
    const int lane    = threadIdx.x & 31;
    const int wave    = threadIdx.x >> 5;
    const int rowBase = blockIdx.x * 128 + wave * 16;
    const bool lowK   = (lane & 16) == 0;  // lanes 0-15 hold K=0,1 ; lanes 16-31 hold K=2,3
    const int lm      = lane & 15;

    // ---- A matrix (16x4 f32): lane L holds row M=L%16, K = 2*(L/16)+v ----
    // Load all 3 components unconditionally, select on values (no exec churn).
    int arow = rowBase + lm;
    if (arow >= n) arow = n - 1;                       // safe clamp (n % 128 == 0 anyway)
    const float* xrow = Xn + (size_t)arow * 3;
    const float x0 = xrow[0], x1 = xrow[1], x2 = xrow[2];
    v2f a;
    a.x = lowK ? x0 : x2;                              // K=0 / K=2
    a.y = lowK ? x1 : 0.0f;                            // K=1 / K=3 (pad)

    float partial[8];
#pragma unroll
    for (int v = 0; v < 8; ++v) partial[v] = 0.0f;

    // ---- 4 column tiles of 16 cover H=64 ----
#pragma unroll
    for (int t = 0; t < 4; ++t) {
        const int col = t * 16 + lm;                   // hidden-unit index for this lane
        // B matrix (4x16 f32): VGPR v, lanes 0-15 -> K=v, lanes 16-31 -> K=v+2.
        // Unconditional loads + v_cndmask selects.
        const float w1k0 = W1[col];
        const float w1k1 = W1[64 + col];
        const float w1k2 = W1[128 + col];
        v2f b;
        b.x = lowK ? w1k0 : w1k2;                      // K=0 / K=2
        b.y = lowK ? w1k1 : 0.0f;                      // K=1 / K=3 (pad)

        v8f c = {};
        v8f d = __builtin_amdgcn_wmma_f32_16x16x4_f32(
            /*neg_a=*/false, a, /*neg_b=*/false, b,
            /*c_mod=*/(short)0, c, /*reuse_a=*/false, /*reuse_b=*/false);

        const float bb = b1[col];
        const float w2 = W2[col];
        // D layout: VGPR v, lane L -> M = v + 8*(L/16), N = L%16 (+16t)
#pragma unroll
        for (int v = 0; v < 8; ++v)
            partial[v] += fast_tanhf(d[v] + bb) * w2;
    }

    // ---- reduce h@W2 across the 16 lanes that share the same row group ----
#pragma unroll
    for (int mask = 1; mask <= 8; mask <<= 1) {
#pragma unroll
        for (int v = 0; v < 8; ++v)
            partial[v] += __shfl_xor(partial[v], mask, 32);
    }

    // lane lm<8 of each half writes row M = lm + 8*(lane>=16)
    if (lm < 8) {
        const int row = rowBase + (lowK ? 0 : 8) + lm;
        if (row < n) {
            const float o = partial[lm] + b2[0];
            ovi_out[row]     = o;
            ws4[4 * row + 0] = o;
        }
    }

    // ---- elementwise per-month mortality/development params ----
    const int prow = blockIdx.x * 128 + (int)threadIdx.x;
    if (threadIdx.x < 128 && prow < n) {
        const float T  = Xr[(size_t)prow * 3 + 0];
        const float u  = (T - 27.0f) * (1.0f / 9.0f);
        const float dp = fmaxf(0.08f * __expf(-u * u), 0.005f);
        const float ma = 0.03f + 0.002f * (T - 26.0f) * (T - 26.0f);
        const float mp = 0.05f + 0.003f * (T - 22.0f) * (T - 22.0f);
        ws4[4 * prow + 1] = dp;
        ws4[4 * prow + 2] = ma;
        ws4[4 * prow + 3] = mp;
    }
}

// One Euler day-step. Pure fma/min/max chain, no branches.
__device__ __forceinline__ void day_step(float& P, float& A,
                                         float ovi, float dpm, float mam,
                                         float mpm, float sdp, float invK) {
    const float density = fmaf(P, invK, 1.0f);
    const float t       = fmaf(mpm, density, dpm);          // dpm + mpm*density
    float dP            = fmaf(-t, P, ovi * A);             // ovi*A - t*P
    float dA            = fmaf(sdp, P, -(mam * A));         // sdp*P - mam*A
    const float maxdP   = fmaf(0.3f, P, 100.0f);
    const float maxdA   = fmaf(0.3f, A, 100.0f);
    dP = fminf(fmaxf(dP, -maxdP), maxdP);
    dA = fminf(fmaxf(dA, -maxdA), maxdA);
    P = fmaxf(P + dP, 1.0f);
    A = fmaxf(A + dA, 1.0f);
}

// ---------------------------------------------------------------------------
// Kernel 2: the irreducible sequential scan. One wave32.
// Lanes cooperatively stream month params (1x global_load_b128 per lane per
// 32-month chunk, register double-buffered), broadcast via __shfl.
// Results saved per-lane, stored coalesced per chunk.
// ---------------------------------------------------------------------------
__global__ void __launch_bounds__(32)
mosq_scan_kernel(const float* __restrict__ ws4,
                 const float* __restrict__ log_sigma,
                 const float* __restrict__ log_K,
                 const float* __restrict__ log_P0,
                 const float* __restrict__ log_A0,
                 const int*   __restrict__ days_ptr,
                 float* __restrict__ A_out,
                 float* __restrict__ P_out,
                 int n)
{
    const int lane = threadIdx.x & 31;

    const float sigma = __builtin_amdgcn_rcpf(1.0f + __expf(-log_sigma[0]));
    const float K     = __expf(log_K[0]) * 1000000.0f;
    const float invK  = 1.0f / K;
    const int   days  = days_ptr[0];

    float P = __expf(log_P0[0]) * 1000.0f;
    float A = __expf(log_A0[0]) * 100.0f;

    const float4* w = (const float4*)ws4;
    const int nChunks = (n + 31) >> 5;

    const int idx0 = lane < n ? lane : (n - 1);
    float4 cur = w[idx0];
    float Asave = 0.0f, Psave = 0.0f;

#pragma unroll 1
    for (int c = 0; c < nChunks; ++c) {
        // prefetch next chunk while this chunk's 32x15 day-steps run
        const int nidx = (c + 1) * 32 + lane;
        float4 nxt = w[(nidx < n) ? nidx : (n - 1)];

        const int mCnt = (n - c * 32) < 32 ? (n - c * 32) : 32;
#pragma unroll 1
        for (int i = 0; i < mCnt; ++i) {
            const float ovi = __shfl(cur.x, i, 32);
            const float dpm = __shfl(cur.y, i, 32);
            const float mam = __shfl(cur.z, i, 32);
            const float mpm = __shfl(cur.w, i, 32);
            const float sdp = sigma * dpm;

            if (days == 15) {
                // common case: fully unrolled dependency chain, zero loop overhead
#pragma unroll
                for (int d = 0; d < 15; ++d)
                    day_step(P, A, ovi, dpm, mam, mpm, sdp, invK);
            } else {
#pragma unroll 1
                for (int d = 0; d < days; ++d)
                    day_step(P, A, ovi, dpm, mam, mpm, sdp, invK);
            }
            if (lane == i) { Asave = A; Psave = P; }   // this lane owns month c*32+i
        }

        const int mrow = c * 32 + lane;
        if (mrow < n) { A_out[mrow] = Asave; P_out[mrow] = Psave; }
        cur = nxt;
    }
}

// ---------------------------------------------------------------------------
extern "C" void kernel_launch(void* const* d_in, const int* in_sizes, int n_in,
                              void* d_out, int out_size, void* d_ws, size_t ws_size,
                              hipStream_t stream) {
    const float* Xn = (const float*)d_in[0];   // weather_norm (n,3)
    const float* Xr = (const float*)d_in[1];   // weather_raw  (n,3)
    const float* W1 = (const float*)d_in[2];   // (3,64)
    const float* b1 = (const float*)d_in[3];   // (64)
    const float* W2 = (const float*)d_in[4];   // (64,1)
    const float* b2 = (const float*)d_in[5];   // (1)
    const float* ls = (const float*)d_in[6];   // log_sigma
    const float* lK = (const float*)d_in[7];   // log_K
    const float* lP = (const float*)d_in[8];   // log_P0_scale
    const float* lA = (const float*)d_in[9];   // log_A0_scale
    const int* days = (const int*)d_in[10];    // days_per_step

    const int n = in_sizes[0] / 3;             // 8192

    float* out     = (float*)d_out;
    float* A_out   = out;                      // A_series
    float* P_out   = out + n;                  // P_series
    float* ovi_out = out + 2 * n;              // ovi
    float* ws4     = (float*)d_ws;             // (n,4) floats = 128 KB

    dim3 block(256);
    dim3 grid((n + 127) / 128);
    mosq_mlp_params_kernel<<<grid, block, 0, stream>>>(
        Xn, Xr, W1, b1, W2, b2, ovi_out, ws4, n);

    mosq_scan_kernel<<<1, 32, 0, stream>>>(
        ws4, ls, lK, lP, lA, days, A_out, P_out, n);
}